// GATNet_68075231642026
// MI455X (gfx1250) — compile-verified
//
#include <hip/hip_runtime.h>
#include <hip/hip_bf16.h>
#include <math.h>

#define NN 50000
#define EE 800000
#define EP (EE + NN)           // edges incl. self-loops = 850000
#define INC 256
#define HID 32
#define HEADS 4
#define L1C (HEADS * HID)      // 128
#define OUTC 16
#define NEG_SLOPE 0.2f
#define EPSV 1e-16f

typedef __attribute__((ext_vector_type(2))) float v2f;
typedef __attribute__((ext_vector_type(4))) float v4f;
typedef __attribute__((ext_vector_type(8))) float v8f;

// Ordered-bits float atomic max: works for any sign mix, init with -inf.
__device__ __forceinline__ void atomicMaxF(float* addr, float val) {
  if (!(val < 0.0f)) atomicMax((int*)addr, __float_as_int(val));
  else               atomicMin((unsigned int*)addr, (unsigned int)__float_as_int(val));
}

// ---------------------------------------------------------------- fill
__global__ void fill_kernel(float* __restrict__ p, float v, int n) {
  int i = blockIdx.x * blockDim.x + threadIdx.x;
  if (i < n) p[i] = v;
}

// ---------------------------------------------------------------- GEMM1: xl1 = x @ W1  (fp32 WMMA 16x16x4)
// grid = NN/16 blocks, 256 threads = 8 waves; wave w owns output cols [16w,16w+16).
__global__ __launch_bounds__(256) void gemm1_wmma(const float* __restrict__ x,
                                                  const float* __restrict__ W1,
                                                  float* __restrict__ xl1) {
  __shared__ float As[16 * INC];   // 16 KB: 16 rows of x
  __shared__ float Bs[64 * L1C];   // 32 KB: 64-deep K-chunk of W1
  const int tid = threadIdx.x;
  const int m0 = blockIdx.x * 16;
  for (int i = tid; i < 16 * INC; i += 256)
    As[i] = x[(size_t)(m0 + (i >> 8)) * INC + (i & 255)];
  const int wave = tid >> 5, lane = tid & 31;
  const int col  = lane & 15;           // A-row M / B-col N / D-col N
  const int kh   = (lane >> 4) << 1;    // lanes 16-31 hold K=2,3
  const int col0 = wave * 16;
  v8f acc = {};
  for (int kc = 0; kc < INC; kc += 64) {
    __syncthreads();
    for (int i = tid; i < 64 * L1C; i += 256)
      Bs[i] = W1[(size_t)(kc + (i >> 7)) * L1C + (i & 127)];
    __syncthreads();
    for (int k = 0; k < 64; k += 4) {
      v2f a, b;
      a.x = As[col * INC + kc + k + kh];
      a.y = As[col * INC + kc + k + kh + 1];
      b.x = Bs[(k + kh) * L1C + col0 + col];
      b.y = Bs[(k + kh + 1) * L1C + col0 + col];
      acc = __builtin_amdgcn_wmma_f32_16x16x4_f32(false, a, false, b,
                                                  (short)0, acc, false, false);
    }
  }
#pragma unroll
  for (int r = 0; r < 8; ++r) {
    int mr = m0 + ((lane < 16) ? r : r + 8);
    xl1[(size_t)mr * L1C + col0 + col] = acc[r];
  }
}

// ---------------------------------------------------------------- GEMM2: xl2 = h @ W2  (fp32 WMMA 16x16x4)
// 8 waves / block, each wave one 16-row tile of the single 16-col output slab.
__global__ __launch_bounds__(256) void gemm2_wmma(const float* __restrict__ h,
                                                  const float* __restrict__ W2,
                                                  float* __restrict__ xl2) {
  __shared__ float Bs[L1C * OUTC];  // 8 KB
  const int tid = threadIdx.x;
  for (int i = tid; i < L1C * OUTC; i += 256) Bs[i] = W2[i];
  __syncthreads();
  const int wave = tid >> 5, lane = tid & 31;
  const int tile = blockIdx.x * 8 + wave;
  if (tile >= NN / 16) return;      // wave-uniform: EXEC stays all-ones for WMMA
  const int m0  = tile * 16;
  const int col = lane & 15;
  const int kh  = (lane >> 4) << 1;
  v8f acc = {};
  for (int k = 0; k < L1C; k += 4) {
    v2f a, b;
    a.x = h[(size_t)(m0 + col) * L1C + k + kh];
    a.y = h[(size_t)(m0 + col) * L1C + k + kh + 1];
    b.x = Bs[(k + kh) * OUTC + col];
    b.y = Bs[(k + kh + 1) * OUTC + col];
    acc = __builtin_amdgcn_wmma_f32_16x16x4_f32(false, a, false, b,
                                                (short)0, acc, false, false);
  }
#pragma unroll
  for (int r = 0; r < 8; ++r) {
    int mr = m0 + ((lane < 16) ? r : r + 8);
    xl2[(size_t)mr * OUTC + col] = acc[r];
  }
}

// ---------------------------------------------------------------- attention coefficients, layer 1
__global__ void alpha1_kernel(const float* __restrict__ xl1,
                              const float* __restrict__ a_src,
                              const float* __restrict__ a_dst,
                              float* __restrict__ as1, float* __restrict__ ad1) {
  int i = blockIdx.x * blockDim.x + threadIdx.x;   // n*HEADS + h
  if (i >= NN * HEADS) return;
  int hd = i & (HEADS - 1);
  const float* v = xl1 + (size_t)(i >> 2) * L1C + hd * HID;
  float ss = 0.f, sd = 0.f;
#pragma unroll 8
  for (int c = 0; c < HID; ++c) {
    float t = v[c];
    ss += t * a_src[hd * HID + c];
    sd += t * a_dst[hd * HID + c];
  }
  as1[i] = ss; ad1[i] = sd;
}

// ---------------------------------------------------------------- edge pass 1 (layer 1): logits + segment max
__global__ void edge1_max(const int* __restrict__ ei, const float* __restrict__ as1,
                          const float* __restrict__ ad1, float* __restrict__ e1,
                          float* __restrict__ m1) {
  int t = blockIdx.x * blockDim.x + threadIdx.x;
  if (t >= EP * HEADS) return;
  int e = t >> 2, hd = t & 3;
  int src = (e < EE) ? ei[e]      : (e - EE);
  int dst = (e < EE) ? ei[EE + e] : (e - EE);
  __builtin_prefetch(ei + e + 8192, 0, 0);         // global_prefetch_b8
  float v = as1[src * HEADS + hd] + ad1[dst * HEADS + hd];
  v = (v > 0.0f) ? v : NEG_SLOPE * v;
  e1[t] = v;
  atomicMaxF(&m1[dst * HEADS + hd], v);
}

// ---------------------------------------------------------------- edge pass 2 (layer 1): exp + segment sum
__global__ void edge1_exp(const int* __restrict__ ei, float* __restrict__ e1,
                          const float* __restrict__ m1, float* __restrict__ dn1) {
  int t = blockIdx.x * blockDim.x + threadIdx.x;
  if (t >= EP * HEADS) return;
  int e = t >> 2, hd = t & 3;
  int dst = (e < EE) ? ei[EE + e] : (e - EE);
  float ex = expf(e1[t] - m1[dst * HEADS + hd]);
  e1[t] = ex;
  atomicAdd(&dn1[dst * HEADS + hd], ex);
}

// ---------------------------------------------------------------- edge pass 3 (layer 1): scatter messages
// one wave per edge; lane owns 4 contiguous channels (head = lane>>3)
__global__ __launch_bounds__(256) void edge1_agg(const int* __restrict__ ei,
                                                 const float* __restrict__ xl1,
                                                 const float* __restrict__ ex1,
                                                 const float* __restrict__ dn1,
                                                 float* __restrict__ agg) {
  int e = blockIdx.x * 8 + (int)(threadIdx.x >> 5);
  if (e >= EP) return;
  int lane = threadIdx.x & 31;
  int src = (e < EE) ? ei[e]      : (e - EE);
  int dst = (e < EE) ? ei[EE + e] : (e - EE);
  int hd = lane >> 3;
  float alpha = ex1[e * HEADS + hd] / (dn1[dst * HEADS + hd] + EPSV);
  const v4f xv = *(const v4f*)(xl1 + (size_t)src * L1C + lane * 4);
  float* o = agg + (size_t)dst * L1C + lane * 4;
  atomicAdd(o + 0, xv.x * alpha);
  atomicAdd(o + 1, xv.y * alpha);
  atomicAdd(o + 2, xv.z * alpha);
  atomicAdd(o + 3, xv.w * alpha);
}

// ---------------------------------------------------------------- bias + ELU (in place)
__global__ void bias_elu(float* __restrict__ hb, const float* __restrict__ b1) {
  int i = blockIdx.x * blockDim.x + threadIdx.x;
  if (i >= NN * L1C) return;
  float v = hb[i] + b1[i & (L1C - 1)];
  hb[i] = (v > 0.0f) ? v : expm1f(v);
}

// ---------------------------------------------------------------- attention coefficients, layer 2
__global__ void alpha2_kernel(const float* __restrict__ xl2,
                              const float* __restrict__ a_src,
                              const float* __restrict__ a_dst,
                              float* __restrict__ as2, float* __restrict__ ad2) {
  int n = blockIdx.x * blockDim.x + threadIdx.x;
  if (n >= NN) return;
  float ss = 0.f, sd = 0.f;
#pragma unroll
  for (int c = 0; c < OUTC; ++c) {
    float t = xl2[(size_t)n * OUTC + c];
    ss += t * a_src[c];
    sd += t * a_dst[c];
  }
  as2[n] = ss; ad2[n] = sd;
}

__global__ void edge2_max(const int* __restrict__ ei, const float* __restrict__ as2,
                          const float* __restrict__ ad2, float* __restrict__ e2,
                          float* __restrict__ m2) {
  int e = blockIdx.x * blockDim.x + threadIdx.x;
  if (e >= EP) return;
  int src = (e < EE) ? ei[e]      : (e - EE);
  int dst = (e < EE) ? ei[EE + e] : (e - EE);
  float v = as2[src] + ad2[dst];
  v = (v > 0.0f) ? v : NEG_SLOPE * v;
  e2[e] = v;
  atomicMaxF(&m2[dst], v);
}

__global__ void edge2_exp(const int* __restrict__ ei, float* __restrict__ e2,
                          const float* __restrict__ m2, float* __restrict__ dn2) {
  int e = blockIdx.x * blockDim.x + threadIdx.x;
  if (e >= EP) return;
  int dst = (e < EE) ? ei[EE + e] : (e - EE);
  float ex = expf(e2[e] - m2[dst]);
  e2[e] = ex;
  atomicAdd(&dn2[dst], ex);
}

__global__ void edge2_agg(const int* __restrict__ ei, const float* __restrict__ xl2,
                          const float* __restrict__ ex2, const float* __restrict__ dn2,
                          float* __restrict__ agg2) {
  int t = blockIdx.x * blockDim.x + threadIdx.x;
  if (t >= EP * 4) return;
  int e = t >> 2, q = t & 3;
  int src = (e < EE) ? ei[e]      : (e - EE);
  int dst = (e < EE) ? ei[EE + e] : (e - EE);
  float alpha = ex2[e] / (dn2[dst] + EPSV);
  const v4f xv = *(const v4f*)(xl2 + (size_t)src * OUTC + q * 4);
  float* o = agg2 + (size_t)dst * OUTC + q * 4;
  atomicAdd(o + 0, xv.x * alpha);
  atomicAdd(o + 1, xv.y * alpha);
  atomicAdd(o + 2, xv.z * alpha);
  atomicAdd(o + 3, xv.w * alpha);
}

// ---------------------------------------------------------------- bias + log_softmax
__global__ void logsoftmax_kernel(const float* __restrict__ agg2,
                                  const float* __restrict__ b2,
                                  float* __restrict__ out) {
  int n = blockIdx.x * blockDim.x + threadIdx.x;
  if (n >= NN) return;
  float v[OUTC];
  float mx = -__builtin_huge_valf();
#pragma unroll
  for (int c = 0; c < OUTC; ++c) {
    v[c] = agg2[(size_t)n * OUTC + c] + b2[c];
    mx = fmaxf(mx, v[c]);
  }
  float s = 0.f;
#pragma unroll
  for (int c = 0; c < OUTC; ++c) s += expf(v[c] - mx);
  float l = logf(s);
#pragma unroll
  for (int c = 0; c < OUTC; ++c) out[(size_t)n * OUTC + c] = v[c] - mx - l;
}

// ================================================================ launch
extern "C" void kernel_launch(void* const* d_in, const int* in_sizes, int n_in,
                              void* d_out, int out_size, void* d_ws, size_t ws_size,
                              hipStream_t stream) {
  const float* x      = (const float*)d_in[0];
  const int*   ei     = (const int*)  d_in[1];   // [2,E] row-major: src then dst
  const float* W1     = (const float*)d_in[2];
  const float* a_src1 = (const float*)d_in[3];
  const float* a_dst1 = (const float*)d_in[4];
  const float* b1     = (const float*)d_in[5];
  const float* W2     = (const float*)d_in[6];
  const float* a_src2 = (const float*)d_in[7];
  const float* a_dst2 = (const float*)d_in[8];
  const float* b2     = (const float*)d_in[9];
  float* out = (float*)d_out;

  // workspace carve-out (~79 MB of floats, all 16-float aligned)
  float* ws = (float*)d_ws;
  size_t off = 0;
  float* xl1  = ws + off; off += (size_t)NN * L1C;    // layer-1 features per node
  float* ex1  = ws + off; off += (size_t)EP * HEADS;  // per-edge logits -> exp
  float* hb   = ws + off; off += (size_t)NN * L1C;    // aggregation -> h after ELU
  float* as1  = ws + off; off += (size_t)NN * HEADS;
  float* ad1  = ws + off; off += (size_t)NN * HEADS;
  float* m1   = ws + off; off += (size_t)NN * HEADS;
  float* dn1  = ws + off; off += (size_t)NN * HEADS;
  float* xl2  = ws + off; off += (size_t)NN * OUTC;
  float* agg2 = ws + off; off += (size_t)NN * OUTC;
  float* as2  = ws + off; off += (size_t)NN;
  float* ad2  = ws + off; off += (size_t)NN;
  float* m2   = ws + off; off += (size_t)NN;
  float* dn2  = ws + off; off += (size_t)NN;
  float* ex2  = ws + off; off += (size_t)EP;
  (void)in_sizes; (void)n_in; (void)out_size; (void)ws_size;

  auto cdiv = [](int a, int b) { return (a + b - 1) / b; };
  const float NEG_INF = -__builtin_huge_valf();

  // init (ws is poisoned; must re-init every call for determinism)
  fill_kernel<<<cdiv(NN * HEADS, 256), 256, 0, stream>>>(m1, NEG_INF, NN * HEADS);
  fill_kernel<<<cdiv(NN * HEADS, 256), 256, 0, stream>>>(dn1, 0.f, NN * HEADS);
  fill_kernel<<<cdiv(NN * L1C, 256), 256, 0, stream>>>(hb, 0.f, NN * L1C);
  fill_kernel<<<cdiv(NN, 256), 256, 0, stream>>>(m2, NEG_INF, NN);
  fill_kernel<<<cdiv(NN, 256), 256, 0, stream>>>(dn2, 0.f, NN);
  fill_kernel<<<cdiv(NN * OUTC, 256), 256, 0, stream>>>(agg2, 0.f, NN * OUTC);

  // ---- layer 1 ----
  gemm1_wmma<<<NN / 16, 256, 0, stream>>>(x, W1, xl1);
  alpha1_kernel<<<cdiv(NN * HEADS, 256), 256, 0, stream>>>(xl1, a_src1, a_dst1, as1, ad1);
  edge1_max<<<cdiv(EP * HEADS, 256), 256, 0, stream>>>(ei, as1, ad1, ex1, m1);
  edge1_exp<<<cdiv(EP * HEADS, 256), 256, 0, stream>>>(ei, ex1, m1, dn1);
  edge1_agg<<<cdiv(EP, 8), 256, 0, stream>>>(ei, xl1, ex1, dn1, hb);
  bias_elu<<<cdiv(NN * L1C, 256), 256, 0, stream>>>(hb, b1);

  // ---- layer 2 ----
  gemm2_wmma<<<cdiv(NN / 16, 8), 256, 0, stream>>>(hb, W2, xl2);
  alpha2_kernel<<<cdiv(NN, 256), 256, 0, stream>>>(xl2, a_src2, a_dst2, as2, ad2);
  edge2_max<<<cdiv(EP, 256), 256, 0, stream>>>(ei, as2, ad2, ex2, m2);
  edge2_exp<<<cdiv(EP, 256), 256, 0, stream>>>(ei, ex2, m2, dn2);
  edge2_agg<<<cdiv(EP * 4, 256), 256, 0, stream>>>(ei, xl2, ex2, dn2, agg2);
  logsoftmax_kernel<<<cdiv(NN, 256), 256, 0, stream>>>(agg2, b2, out);
}